// LuongAttention_17051020165767
// MI455X (gfx1250) — compile-verified
//
#include <hip/hip_runtime.h>
#include <stdint.h>
#include <math.h>

// ---------------- problem constants ----------------
#define B_      64
#define S_      2048
#define H_      1024
#define NSPLIT  16                     // workgroups per batch (S split)
#define ROWS_WG (S_ / NSPLIT)          // 128 rows per WG
#define BLKROWS 16                     // rows per LDS tile (one WMMA M-tile)
#define NBLK    (ROWS_WG / BLKROWS)    // 8 tiles per WG
#define NTHR    256
#define NWAVES  8
#define KSEG    (H_ / NWAVES)          // 128 K per wave
#define PITCH   1028                   // padded row pitch (floats): 1028%64=4 -> conflict-free, 4112B = 257*16
#define SEGROW  (H_ / 4)               // 16B segments per row = 256
#define SEGBLK  (BLKROWS * SEGROW)     // 4096 segments per tile
#define SEGWAVE (SEGBLK / NWAVES)      // 512
#define ASYNC_PER_WAVE (SEGWAVE / 32)  // 16 async b128 per wave per tile

// ---------------- LDS layout (floats) ----------------
#define SM_DEC    0                    // 1024
#define SM_SCORES 1024                 // 16
#define SM_SPART  1040                 // 8 waves * 16
#define SM_BUF0   1168                 // 16B aligned (1168*4 = 292*16)
#define SM_BUFSZ  (BLKROWS * PITCH)    // 16448
#define SM_TOTAL  (SM_BUF0 + 2 * SM_BUFSZ)   // 34064 floats = 136256 B

typedef __attribute__((ext_vector_type(2))) float v2f;
typedef __attribute__((ext_vector_type(8))) float v8f;

// ---- async copy: one 16-row x 1024-float enc tile -> LDS (b128 per lane) ----
__device__ __forceinline__ void issue_tile_async(const float* __restrict__ encb,
                                                 float* smem, int buf,
                                                 int s_first, int wave, int lane) {
    uint32_t lds_base = (uint32_t)(uintptr_t)(smem + SM_BUF0 + buf * SM_BUFSZ);
    uint64_t gbase    = (uint64_t)(uintptr_t)encb;
#pragma unroll
    for (int it = 0; it < ASYNC_PER_WAVE; ++it) {
        int seg = wave * SEGWAVE + it * 32 + lane;   // linear 16B segment id
        int row = seg >> 8;                          // /SEGROW
        int c16 = seg & 255;
        uint32_t voff  = (uint32_t)((s_first + row) * (H_ * 4) + c16 * 16);
        uint32_t laddr = lds_base + (uint32_t)(row * (PITCH * 4) + c16 * 16);
        asm volatile("global_load_async_to_lds_b128 %0, %1, %2"
                     :: "v"(laddr), "v"(voff), "s"(gbase) : "memory");
    }
}

// ================= pass 1: fused scores + online softmax + partial context ===
__global__ __launch_bounds__(NTHR)
void luong_pass1(const float* __restrict__ hdec, const float* __restrict__ cdec,
                 const float* __restrict__ enc,
                 float* __restrict__ ws_scores,   // [B,S] raw scores
                 float* __restrict__ ws_ml,       // [B*NSPLIT][2] partial m,l
                 float* __restrict__ ws_ctx) {    // [B*NSPLIT][H] partial ctx
    extern __shared__ float smem[];
    const int tid   = threadIdx.x;
    const int lane  = tid & 31;
    const int wave  = tid >> 5;
    const int split = blockIdx.x;
    const int b     = blockIdx.y;
    const int s0    = split * ROWS_WG;
    const float* encb = enc + (size_t)b * S_ * H_;

    // dec = h + c -> LDS (256 thr * 4 floats)
    {
        const int h0 = tid * 4;
        float4 hv = *(const float4*)(hdec + (size_t)b * H_ + h0);
        float4 cv = *(const float4*)(cdec + (size_t)b * H_ + h0);
        float4 d; d.x = hv.x + cv.x; d.y = hv.y + cv.y;
                  d.z = hv.z + cv.z; d.w = hv.w + cv.w;
        *(float4*)(smem + SM_DEC + h0) = d;
    }

    // prefetch tile 0
    issue_tile_async(encb, smem, 0, s0, wave, lane);

    float m = -INFINITY, l = 0.f;
    float c0 = 0.f, c1 = 0.f, c2 = 0.f, c3 = 0.f;
    const int h0 = tid * 4;

    __syncthreads();   // dec ready

    for (int blk = 0; blk < NBLK; ++blk) {
        const int buf = blk & 1;
        // prefetch next tile into other buffer, then wait for current tile
        if (blk + 1 < NBLK) {
            issue_tile_async(encb, smem, buf ^ 1, s0 + (blk + 1) * BLKROWS, wave, lane);
            asm volatile("s_wait_asynccnt 16" ::: "memory");
        } else {
            asm volatile("s_wait_asynccnt 0" ::: "memory");
        }
        __syncthreads();   // tile visible to all waves

        const float* tile = smem + SM_BUF0 + buf * SM_BUFSZ;

        // ---- WMMA f32 16x16x4: scores for 16 rows, K segment per wave ----
        {
            const float* arow = tile + (lane & 15) * PITCH;
            const int csel = (lane >> 4) * 2;      // lanes 0-15: K0/K1, 16-31: K2/K3
            v8f acc = {0.f, 0.f, 0.f, 0.f, 0.f, 0.f, 0.f, 0.f};
#pragma unroll
            for (int k = 0; k < KSEG; k += 4) {
                const int col = wave * KSEG + k + csel;
                v2f a  = *(const v2f*)(arow + col);            // A: enc[m, k..k+1]
                v2f bb = *(const v2f*)(smem + SM_DEC + col);   // B: dec broadcast cols
                acc = __builtin_amdgcn_wmma_f32_16x16x4_f32(
                          false, a, false, bb, (short)0, acc, false, false);
            }
            // C layout: VGPR r = row r (lanes 0-15) / row r+8 (lanes 16-31); cols identical
            if ((lane & 15) == 0) {
                const int rbase = (lane >> 4) * 8;
#pragma unroll
                for (int j = 0; j < 8; ++j)
                    smem[SM_SPART + wave * 16 + rbase + j] = acc[j];
            }
        }
        __syncthreads();

        // deterministic cross-wave K reduction; stash raw scores
        if (tid < 16) {
            float sacc = 0.f;
#pragma unroll
            for (int w = 0; w < NWAVES; ++w) sacc += smem[SM_SPART + w * 16 + tid];
            smem[SM_SCORES + tid] = sacc;
            ws_scores[(size_t)b * S_ + s0 + blk * BLKROWS + tid] = sacc;
        }
        __syncthreads();

        // ---- online softmax + context accumulation (m,l replicated per thread) ----
        float blkmax = smem[SM_SCORES + 0];
#pragma unroll
        for (int r = 1; r < 16; ++r) blkmax = fmaxf(blkmax, smem[SM_SCORES + r]);
        const float m_new = fmaxf(m, blkmax);
        const float scale = __expf(m - m_new);
        l *= scale; c0 *= scale; c1 *= scale; c2 *= scale; c3 *= scale;
#pragma unroll
        for (int r = 0; r < 16; ++r) {
            const float p = __expf(smem[SM_SCORES + r] - m_new);
            l += p;
            const float4 e = *(const float4*)(tile + r * PITCH + h0);
            c0 = fmaf(p, e.x, c0); c1 = fmaf(p, e.y, c1);
            c2 = fmaf(p, e.z, c2); c3 = fmaf(p, e.w, c3);
        }
        m = m_new;
        __syncthreads();   // everyone done reading tile before it is refilled
    }

    // write per-split partials
    const size_t pid = (size_t)b * NSPLIT + split;
    float4 cv; cv.x = c0; cv.y = c1; cv.z = c2; cv.w = c3;
    *(float4*)(ws_ctx + pid * H_ + h0) = cv;
    if (tid == 0) { ws_ml[pid * 2] = m; ws_ml[pid * 2 + 1] = l; }
}

// ================= pass 2: final (m,l) per batch =================
__global__ void luong_reduce_ml(const float* __restrict__ ws_ml,
                                float* __restrict__ ws_mlf) {
    const int b = threadIdx.x;
    if (b >= B_) return;
    float m = -INFINITY;
    for (int i = 0; i < NSPLIT; ++i) m = fmaxf(m, ws_ml[(b * NSPLIT + i) * 2]);
    float l = 0.f;
    for (int i = 0; i < NSPLIT; ++i)
        l += __expf(ws_ml[(b * NSPLIT + i) * 2] - m) * ws_ml[(b * NSPLIT + i) * 2 + 1];
    ws_mlf[b * 2] = m; ws_mlf[b * 2 + 1] = l;
}

// ================= pass 3: combine partial contexts -> d_out[0 : B*H) ========
__global__ void luong_combine_ctx(const float* __restrict__ ws_ctx,
                                  const float* __restrict__ ws_ml,
                                  const float* __restrict__ ws_mlf,
                                  float* __restrict__ out) {
    const int b = blockIdx.y;
    const int h = blockIdx.x * blockDim.x + threadIdx.x;
    const float m = ws_mlf[b * 2], l = ws_mlf[b * 2 + 1];
    float acc = 0.f;
    for (int i = 0; i < NSPLIT; ++i) {
        const float w = __expf(ws_ml[(b * NSPLIT + i) * 2] - m);
        acc = fmaf(w, ws_ctx[((size_t)b * NSPLIT + i) * H_ + h], acc);
    }
    out[(size_t)b * H_ + h] = acc / l;
}

// ================= pass 4: attention weights -> d_out[B*H : B*H+B*S) =========
__global__ void luong_attn_out(const float* __restrict__ ws_scores,
                               const float* __restrict__ ws_mlf,
                               float* __restrict__ out) {
    const size_t idx = (size_t)blockIdx.x * blockDim.x + threadIdx.x; // b*S+s
    const int b = (int)(idx >> 11);                                   // S_ = 2048
    out[(size_t)B_ * H_ + idx] =
        __expf(ws_scores[idx] - ws_mlf[b * 2]) / ws_mlf[b * 2 + 1];
}

// ================= launcher =================
extern "C" void kernel_launch(void* const* d_in, const int* in_sizes, int n_in,
                              void* d_out, int out_size, void* d_ws, size_t ws_size,
                              hipStream_t stream) {
    const float* hdec = (const float*)d_in[0];   // [B,H]
    const float* cdec = (const float*)d_in[1];   // [B,H]
    const float* enc  = (const float*)d_in[2];   // [B,S,H]
    float* out = (float*)d_out;                  // [B*H] ctx ++ [B*S] attn

    // workspace layout (floats)
    float* ws        = (float*)d_ws;
    float* ws_scores = ws;                                  // B*S       = 131072
    float* ws_ml     = ws_scores + (size_t)B_ * S_;         // B*NSPLIT*2=   2048
    float* ws_ctx    = ws_ml + (size_t)B_ * NSPLIT * 2;     // B*NSPLIT*H=1048576 (16B aligned)
    float* ws_mlf    = ws_ctx + (size_t)B_ * NSPLIT * H_;   // B*2

    luong_pass1<<<dim3(NSPLIT, B_), NTHR, SM_TOTAL * sizeof(float), stream>>>(
        hdec, cdec, enc, ws_scores, ws_ml, ws_ctx);
    luong_reduce_ml<<<1, B_, 0, stream>>>(ws_ml, ws_mlf);
    luong_combine_ctx<<<dim3(H_ / 256, B_), 256, 0, stream>>>(ws_ctx, ws_ml, ws_mlf, out);
    luong_attn_out<<<(B_ * S_) / 256, 256, 0, stream>>>(ws_scores, ws_mlf, out);
}